// GraphConvolution_83786222010494
// MI455X (gfx1250) — compile-verified
//
#include <hip/hip_runtime.h>
#include <hip/hip_bf16.h>
#include <stdint.h>

#define IN_F  512
#define OUT_F 128
#define KTILE 64
#define NTILES (IN_F / KTILE)   // 8 K-tiles

typedef __attribute__((ext_vector_type(2))) float v2f;
typedef __attribute__((ext_vector_type(8))) float v8f;

// ---------------------------------------------------------------------------
// gfx1250 async Global->LDS copy (ASYNCcnt-tracked, bypasses VGPRs).
// vdst = LDS byte offset (wave-relative), vaddr = 64-bit global address.
// ---------------------------------------------------------------------------
__device__ __forceinline__ void async_copy_b128(unsigned lds_byte_off,
                                                const float* gptr) {
  asm volatile("global_load_async_to_lds_b128 %0, %1, off"
               :: "v"(lds_byte_off), "v"(gptr)
               : "memory");
}

__device__ __forceinline__ void wait_asynccnt0() {
  asm volatile("s_wait_asynccnt 0" ::: "memory");
}

// Truncate a generic pointer to a __shared__ object to its LDS byte offset
// (ISA: flat->LDS mapping uses addr[31:0] as the wave-relative LDS address).
__device__ __forceinline__ unsigned lds_off(const void* p) {
  return (unsigned)(uintptr_t)p;
}

// Stage one K-tile (A: 16xKTILE, B: KTILExOUT_F) into a fixed LDS buffer.
__device__ __forceinline__ void stage_tile(int tid, int m0, int k0,
                                           const float* __restrict__ x,
                                           const float* __restrict__ w,
                                           float* xsb, float* wsbb) {
  const int ar = tid >> 4, ac = (tid & 15) << 2;       // 16 rows x 16 lanes
  async_copy_b128(lds_off(xsb + ar * KTILE + ac),
                  x + (size_t)(m0 + ar) * IN_F + k0 + ac);
#pragma unroll
  for (int j = 0; j < 8; ++j) {
    int i = tid + j * 256;
    int br = i >> 5, bc = (i & 31) << 2;               // 64 rows x 32 f4 cols
    async_copy_b128(lds_off(wsbb + br * OUT_F + bc),
                    w + (size_t)(k0 + br) * OUT_F + bc);
  }
}

// 16 chained V_WMMA_F32_16X16X4_F32 over one staged K-tile.
__device__ __forceinline__ v8f compute_tile(v8f acc, int m, int half, int n,
                                            const float* xsb,
                                            const float* wsbb) {
#pragma unroll
  for (int kk = 0; kk < KTILE; kk += 4) {
    v2f a, b;
    // A 16x4 fp32 layout: VGPR0 = K{0|2}, VGPR1 = K{1|3} by lane half
    a.x = xsb[m * KTILE + kk + 2 * half + 0];
    a.y = xsb[m * KTILE + kk + 2 * half + 1];
    // B 4x16 fp32 layout mirrors A: rows K{0|2}/K{1|3}, cols on lanes
    b.x = wsbb[(kk + 2 * half + 0) * OUT_F + n];
    b.y = wsbb[(kk + 2 * half + 1) * OUT_F + n];
    acc = __builtin_amdgcn_wmma_f32_16x16x4_f32(
        /*neg_a=*/false, a, /*neg_b=*/false, b,
        /*c_mod=*/(short)0, acc, /*reuse_a=*/false, /*reuse_b=*/false);
  }
  return acc;
}

// ---------------------------------------------------------------------------
// Phase 1: support = x @ W.  Block = 256 threads = 8 waves; each block does a
// 16-row x 128-col strip (wave w owns N-tile [16w,16w+16)).  Double-buffered
// async-to-LDS staging, K pipeline unrolled by 2 so each half addresses a
// compile-time-constant LDS buffer (immediate-offset paired ds loads).
// ---------------------------------------------------------------------------
__global__ __launch_bounds__(256) void gcn_gemm_wmma(
    const float* __restrict__ x, const float* __restrict__ w,
    float* __restrict__ support, int n_nodes) {
  __shared__ float xs[2][16 * KTILE];      // 2 x 4 KB  : A strips
  __shared__ float wsb[2][KTILE * OUT_F];  // 2 x 32 KB : B strips

  const int tid  = threadIdx.x;
  const int lane = tid & 31;
  const int wave = tid >> 5;
  const int m0   = blockIdx.x * 16;

  const int m    = lane & 15;
  const int half = lane >> 4;
  const int n    = (wave << 4) + (lane & 15);

  // Prologue: stage tile 0 into buffer 0.
  stage_tile(tid, m0, 0, x, w, xs[0], wsb[0]);
  wait_asynccnt0();
  __syncthreads();

  v8f acc = {};

  for (int t = 0; t < NTILES; t += 2) {
    // Stage tile t+1 -> buf1 while computing tile t from buf0.
    stage_tile(tid, m0, (t + 1) * KTILE, x, w, xs[1], wsb[1]);
    acc = compute_tile(acc, m, half, n, xs[0], wsb[0]);
    wait_asynccnt0();
    __syncthreads();

    // Stage tile t+2 -> buf0 while computing tile t+1 from buf1.
    if (t + 2 < NTILES) {
      stage_tile(tid, m0, (t + 2) * KTILE, x, w, xs[0], wsb[0]);
      acc = compute_tile(acc, m, half, n, xs[1], wsb[1]);
      wait_asynccnt0();
      __syncthreads();
    } else {
      acc = compute_tile(acc, m, half, n, xs[1], wsb[1]);
    }
  }

  // C/D layout: VGPR vi holds row (vi + 8*half), column n.
#pragma unroll
  for (int vi = 0; vi < 8; ++vi) {
    int row = m0 + vi + 8 * half;
    if (row < n_nodes) support[row * OUT_F + n] = acc[vi];
  }
}

// ---------------------------------------------------------------------------
// Phase 2a: zero the accumulator (d_out is poisoned by the harness)
// ---------------------------------------------------------------------------
__global__ void gcn_zero(float* __restrict__ out, int n) {
  int i = blockIdx.x * blockDim.x + threadIdx.x;
  if (i < n) out[i] = 0.0f;
}

// ---------------------------------------------------------------------------
// Phase 2b: out[row[e]] += val[e] * support[col[e]]  (scatter-add)
// 128 threads per edge; 2 edges per 256-thread block. support (25.6 MB) is
// L2-resident (192 MB), so gathers and atomics are L2-side operations.
// ---------------------------------------------------------------------------
__global__ __launch_bounds__(256) void gcn_scatter(
    const int* __restrict__ erow, const int* __restrict__ ecol,
    const float* __restrict__ eval, const float* __restrict__ support,
    float* __restrict__ out, int n_edges) {
  int e = blockIdx.x * 2 + (threadIdx.x >> 7);
  int f = threadIdx.x & 127;
  if (e < n_edges) {
    float v = eval[e];
    float s = support[ecol[e] * OUT_F + f];
    atomicAdd(&out[erow[e] * OUT_F + f], v * s);
  }
}

// ---------------------------------------------------------------------------
// Phase 3: out = selu(out) + bias   (in place, after aggregation completes)
// ---------------------------------------------------------------------------
__global__ void gcn_selu_bias(float* __restrict__ out,
                              const float* __restrict__ bias, int n) {
  int i = blockIdx.x * blockDim.x + threadIdx.x;
  if (i < n) {
    const float scale = 1.0507009873554805f;
    const float alpha = 1.6732632423543772f;
    float v = out[i];
    v = scale * (v > 0.0f ? v : alpha * (expf(v) - 1.0f));
    out[i] = v + bias[i & (OUT_F - 1)];
  }
}

// ---------------------------------------------------------------------------
extern "C" void kernel_launch(void* const* d_in, const int* in_sizes, int n_in,
                              void* d_out, int out_size, void* d_ws, size_t ws_size,
                              hipStream_t stream) {
  const float* x    = (const float*)d_in[0];
  const int*   erow = (const int*)  d_in[1];
  const int*   ecol = (const int*)  d_in[2];
  const float* evalv= (const float*)d_in[3];
  const float* w    = (const float*)d_in[4];
  const float* bias = (const float*)d_in[5];

  const int n_nodes = in_sizes[0] / IN_F;   // 50000
  const int n_edges = in_sizes[1];          // 800000
  const int tot     = n_nodes * OUT_F;

  float* support = (float*)d_ws;            // 50000*128*4 = 25.6 MB scratch
  float* out     = (float*)d_out;

  gcn_gemm_wmma<<<(n_nodes + 15) / 16, 256, 0, stream>>>(x, w, support, n_nodes);
  gcn_zero<<<(tot + 255) / 256, 256, 0, stream>>>(out, tot);
  gcn_scatter<<<(n_edges + 1) / 2, 256, 0, stream>>>(erow, ecol, evalv, support, out, n_edges);
  gcn_selu_bias<<<(tot + 255) / 256, 256, 0, stream>>>(out, bias, tot);
}